// Cost_volume_s2_72645076845119
// MI455X (gfx1250) — compile-verified
//
#include <hip/hip_runtime.h>

// Cost-volume (stereo variance cost), MI455X / gfx1250.
//
// feat_* [4,32,256,256] f32, disp_init [4,1,256,256] f32,
// out [4,8,9,256,256] f32.
//
// Roofline: ~180 MB @ 23.3 TB/s => ~7.7 us floor; no matmul -> no WMMA role.
// CDNA5 path: GLOBAL_LOAD_ASYNC_TO_LDS (ASYNCcnt DMA) stages each (b,h) row
// of feat_ls/feat_rs (32ch x 256px = 32 KB each) into LDS once.
//
// LDS-volume optimization: all 9 disparities' 2 lerp taps per side live in a
// 3-pixel window starting at floor(x +/- disp - 0.4) (span 0.8 < 1). Load
// the window once per (channel, side) into registers (6 B32 instead of 36),
// scatter the per-disparity lerp weights into 3-weight vectors shared across
// channels. Inner body = 9 FMAs per (channel, disparity), VOPD-pairable:
//   d1 = dot3(lwin, WL[d]) - ref ; d2 = dot3(rwin, WR[d]) - ref
//   9*var = 2*(d1^2 + d2^2 - d1*d2) ; cost = sum_k(...) / 18

#define B_   4
#define C_   32
#define H_   256
#define W_   256
#define D_   9
#define G_   8
#define CPG_ 4

typedef int v4i_ __attribute__((vector_size(16)));
typedef __attribute__((address_space(1))) v4i_* as1_v4i;
typedef __attribute__((address_space(3))) v4i_* as3_v4i;

__device__ __forceinline__ void async_copy_b128(const void* g, void* l) {
#if __has_builtin(__builtin_amdgcn_global_load_async_to_lds_b128)
  __builtin_amdgcn_global_load_async_to_lds_b128(
      (as1_v4i)(unsigned long long)g,
      (as3_v4i)(unsigned)(unsigned long long)l,
      /*offset=*/0, /*cpol=*/0);
#else
  unsigned lds_off = (unsigned)(unsigned long long)l;
  asm volatile("global_load_async_to_lds_b128 %0, %1, off"
               :: "v"(lds_off), "v"(g) : "memory");
#endif
}

__device__ __forceinline__ void wait_async_zero() {
#if __has_builtin(__builtin_amdgcn_s_wait_asynccnt)
  __builtin_amdgcn_s_wait_asynccnt(0);
#else
  asm volatile("s_wait_asynccnt 0" ::: "memory");
#endif
}

__global__ __launch_bounds__(256) void cost_volume_kernel(
    const float* __restrict__ feat_ref,
    const float* __restrict__ feat_ls,
    const float* __restrict__ feat_rs,
    const float* __restrict__ disp_init,
    float* __restrict__ out) {
  __shared__ __align__(16) float s_ls[C_ * W_];  // 32 KB
  __shared__ __align__(16) float s_rs[C_ * W_];  // 32 KB

  const int h   = blockIdx.x;
  const int b   = blockIdx.y;
  const int tid = threadIdx.x;   // == pixel x coordinate
  const int w   = tid;

  const size_t plane   = (size_t)H_ * W_;
  const size_t rowbase = (size_t)b * C_ * plane + (size_t)h * W_;

  // ---- Stage feat_ls / feat_rs rows (all 32 channels) into LDS, async. ----
  {
    const int cl = tid >> 6;          // 0..3 : channel sub-index
    const int e  = (tid & 63) << 2;   // float index within the 256-wide row
#pragma unroll
    for (int i = 0; i < 8; ++i) {
      const int c = i * 4 + cl;
      async_copy_b128(feat_ls + rowbase + (size_t)c * plane + e,
                      &s_ls[c * W_ + e]);
      async_copy_b128(feat_rs + rowbase + (size_t)c * plane + e,
                      &s_rs[c * W_ + e]);
    }
  }

  // ---- Overlap with async DMA: taps, weights, feat_ref registers ----
  const float dispv = disp_init[((size_t)b * H_ + h) * W_ + w];

  const float res[D_] = {-0.4f, -0.3f, -0.2f, -0.1f, 0.0f,
                          0.1f,  0.2f,  0.3f,  0.4f};

  // 3-tap window bases: smallest sample x over d (left: d=0, right: d=8).
  const int wbL = (int)floorf((float)w + (dispv + res[0]));
  const int wbR = (int)floorf((float)w - (dispv + res[D_ - 1]));

  // Clamped window tap indices (weights for clamped-away taps are 0).
  int tL[3], tR[3];
#pragma unroll
  for (int j = 0; j < 3; ++j) {
    tL[j] = min(max(wbL + j, 0), W_ - 1);
    tR[j] = min(max(wbR + j, 0), W_ - 1);
  }

  // Per-disparity 3-weight vectors (shared across all channels).
  float WL[D_][3], WR[D_][3];
#pragma unroll
  for (int d = 0; d < D_; ++d) {
    const float dd = dispv + res[d];
    {  // left: sample at x + disp
      const float x  = (float)w + dd;
      const float xf = floorf(x);
      const int   x0 = (int)xf;
      const float fr = x - xf;
      const float w0 = (x0 >= 0 && x0 < W_) ? (1.0f - fr) : 0.0f;
      const float w1 = (x0 + 1 >= 0 && x0 + 1 < W_) ? fr : 0.0f;
      const int   r  = min(max(x0 - wbL, 0), 1);   // window slot of tap0
      WL[d][0] = r ? 0.0f : w0;
      WL[d][1] = r ? w0 : w1;
      WL[d][2] = r ? w1 : 0.0f;
    }
    {  // right: sample at x - disp
      const float x  = (float)w - dd;
      const float xf = floorf(x);
      const int   x0 = (int)xf;
      const float fr = x - xf;
      const float w0 = (x0 >= 0 && x0 < W_) ? (1.0f - fr) : 0.0f;
      const float w1 = (x0 + 1 >= 0 && x0 + 1 < W_) ? fr : 0.0f;
      const int   r  = min(max(x0 - wbR, 0), 1);
      WR[d][0] = r ? 0.0f : w0;
      WR[d][1] = r ? w0 : w1;
      WR[d][2] = r ? w1 : 0.0f;
    }
  }

  // feat_ref is never warped: this thread only needs its own column.
  float frv[C_];
#pragma unroll
  for (int c = 0; c < C_; ++c)
    frv[c] = feat_ref[rowbase + (size_t)c * plane + w];

  wait_async_zero();   // s_wait_asynccnt 0 : my wave's DMA landed in LDS
  __syncthreads();     // everyone's DMA landed

  // ---- Compute ----
#pragma unroll
  for (int g = 0; g < G_; ++g) {
    // Load 3-tap windows for this group's 4 channels, both sides (24 B32).
    float lw[3][CPG_], rw[3][CPG_];
#pragma unroll
    for (int k = 0; k < CPG_; ++k) {
      const int c = g * CPG_ + k;
#pragma unroll
      for (int j = 0; j < 3; ++j) {
        lw[j][k] = s_ls[c * W_ + tL[j]];
        rw[j][k] = s_rs[c * W_ + tR[j]];
      }
    }
#pragma unroll
    for (int d = 0; d < D_; ++d) {
      float acc = 0.0f;
#pragma unroll
      for (int k = 0; k < CPG_; ++k) {
        const float fv = frv[g * CPG_ + k];
        const float d1 = fmaf(lw[0][k], WL[d][0],
                         fmaf(lw[1][k], WL[d][1],
                         fmaf(lw[2][k], WL[d][2], -fv)));
        const float d2 = fmaf(rw[0][k], WR[d][0],
                         fmaf(rw[1][k], WR[d][1],
                         fmaf(rw[2][k], WR[d][2], -fv)));
        acc = fmaf(d1, d1, acc);
        acc = fmaf(d2, d2, acc);
        acc = fmaf(d1, -d2, acc);
      }
      const float cost = acc * (1.0f / 18.0f);
      __builtin_nontemporal_store(
          cost, &out[((((size_t)b * G_ + g) * D_ + d) * H_ + h) * W_ + w]);
    }
  }
}

extern "C" void kernel_launch(void* const* d_in, const int* in_sizes, int n_in,
                              void* d_out, int out_size, void* d_ws,
                              size_t ws_size, hipStream_t stream) {
  (void)in_sizes; (void)n_in; (void)out_size; (void)d_ws; (void)ws_size;
  const float* feat_ref  = (const float*)d_in[0];
  const float* feat_ls   = (const float*)d_in[1];
  const float* feat_rs   = (const float*)d_in[2];
  const float* disp_init = (const float*)d_in[3];
  float* out = (float*)d_out;

  dim3 grid(H_, B_);   // one workgroup per (batch, image-row)
  cost_volume_kernel<<<grid, 256, 0, stream>>>(feat_ref, feat_ls, feat_rs,
                                               disp_init, out);
}